// MultiHuberLoss_62036507623929
// MI455X (gfx1250) — compile-verified
//
#include <hip/hip_runtime.h>

// MultiHuberLoss on MI455X (gfx1250, wave32).
// Memory-bound: 262 MB streamed once -> ~11.3 us floor @ 23.3 TB/s.
// Key identity: treat every element as non-target (m = -x), then add a single
// per-row correction for the target column. This makes the streaming pass
// completely flat: no row indexing, no divergence, exact trip counts.
//
//   loss_nontarget(x) = (x <= 1) ? max(0, 1+x)^2 : 4x
//   loss_target(x)    = (x >= -1) ? max(0, 1-x)^2 : -4x
//
// Pass 1: flat stream of 16,384,000 float4s, NT b128 loads, 2560 blocks x 256,
//         exactly 25 iters/thread (no bounds checks). Block partial -> d_ws.
// Pass 2: 65,536 target corrections (one gather per row), 256 blocks x 256.
// Pass 3: deterministic single-block f64 reduce of 2816 partials -> sum/N.

typedef float v4f __attribute__((ext_vector_type(4)));

#define NROWS   65536
#define CCOLS   1000
#define NV4     16384000                 // 65536 * 250
#define TPB     256
#define WPB     (TPB / 32)               // 8 waves per block
#define SBLK    2560                     // streaming blocks
#define STHREADS (SBLK * TPB)            // 655360
#define ITERS   (NV4 / STHREADS)         // exactly 25
#define CBLK    (NROWS / TPB)            // 256 correction blocks
#define NPART   (SBLK + CBLK)            // 2816 partials

__device__ __forceinline__ float loss_nontarget(float x) {
    float h   = fmaxf(0.0f, 1.0f + x);
    float q   = h * h;
    float lin = 4.0f * x;
    return (x <= 1.0f) ? q : lin;
}

__device__ __forceinline__ float loss_target(float x) {
    float h   = fmaxf(0.0f, 1.0f - x);
    float q   = h * h;
    float lin = -4.0f * x;
    return (x >= -1.0f) ? q : lin;
}

// wave32 butterfly + cross-wave LDS reduce; lane0/wave0 writes the block sum
__device__ __forceinline__ void block_reduce_store(float acc, float* dst) {
    const int lane = threadIdx.x & 31;
    const int wid  = threadIdx.x >> 5;
    #pragma unroll
    for (int off = 16; off > 0; off >>= 1)
        acc += __shfl_xor(acc, off, 32);
    __shared__ float smem[WPB];
    if (lane == 0) smem[wid] = acc;
    __syncthreads();
    if (wid == 0) {
        float b = (lane < WPB) ? smem[lane] : 0.0f;
        #pragma unroll
        for (int off = 16; off > 0; off >>= 1)
            b += __shfl_xor(b, off, 32);
        if (lane == 0) *dst = b;
    }
}

__global__ void __launch_bounds__(TPB)
huber_stream_kernel(const float* __restrict__ inp, float* __restrict__ partial) {
    const v4f* __restrict__ p = (const v4f*)inp;
    const int gid = blockIdx.x * TPB + threadIdx.x;

    float acc = 0.0f;
    #pragma unroll 5
    for (int i = 0; i < ITERS; ++i) {
        // contiguous, fully coalesced, exactly divisible: no bounds check
        v4f v = __builtin_nontemporal_load(p + ((size_t)i * STHREADS + gid));
        acc += loss_nontarget(v.x);
        acc += loss_nontarget(v.y);
        acc += loss_nontarget(v.z);
        acc += loss_nontarget(v.w);
    }
    block_reduce_store(acc, &partial[blockIdx.x]);
}

__global__ void __launch_bounds__(TPB)
huber_correction_kernel(const float* __restrict__ inp,
                        const long long* __restrict__ tgt,
                        float* __restrict__ partial) {
    const int row = blockIdx.x * TPB + threadIdx.x;   // exactly covers [0, 65536)
    const int t   = (int)tgt[row];                    // value in [0, 1000)
    const float x = inp[(size_t)row * CCOLS + t];
    float acc = loss_target(x) - loss_nontarget(x);
    block_reduce_store(acc, &partial[SBLK + blockIdx.x]);
}

__global__ void __launch_bounds__(TPB)
huber_final_kernel(const float* __restrict__ partial, float* __restrict__ out) {
    const int lane = threadIdx.x & 31;
    const int wid  = threadIdx.x >> 5;

    double acc = 0.0;
    for (int i = threadIdx.x; i < NPART; i += TPB)
        acc += (double)partial[i];

    #pragma unroll
    for (int off = 16; off > 0; off >>= 1)
        acc += __shfl_xor(acc, off, 32);

    __shared__ double smem[WPB];
    if (lane == 0) smem[wid] = acc;
    __syncthreads();

    if (wid == 0) {
        double b = (lane < WPB) ? smem[lane] : 0.0;
        #pragma unroll
        for (int off = 16; off > 0; off >>= 1)
            b += __shfl_xor(b, off, 32);
        if (lane == 0)
            out[0] = (float)(b / (double)NROWS);
    }
}

extern "C" void kernel_launch(void* const* d_in, const int* in_sizes, int n_in,
                              void* d_out, int out_size, void* d_ws, size_t ws_size,
                              hipStream_t stream) {
    const float*     inp = (const float*)d_in[0];
    const long long* tgt = (const long long*)d_in[1];
    float*           out = (float*)d_out;
    float*           partial = (float*)d_ws;   // NPART * 4 B = 11.3 KB scratch

    huber_stream_kernel    <<<SBLK, TPB, 0, stream>>>(inp, partial);
    huber_correction_kernel<<<CBLK, TPB, 0, stream>>>(inp, tgt, partial);
    huber_final_kernel     <<<1,    TPB, 0, stream>>>(partial, out);
}